// DownConvLayers_46531675685215
// MI455X (gfx1250) — compile-verified
//
#include <hip/hip_runtime.h>

typedef float v2f __attribute__((ext_vector_type(2)));
typedef float v8f __attribute__((ext_vector_type(8)));

// ---------------------------------------------------------------------------
// Degree / normalization kernels
// ---------------------------------------------------------------------------
__global__ void k_init_deg(float* __restrict__ deg, int N) {
    int i = blockIdx.x * blockDim.x + threadIdx.x;
    if (i < N) deg[i] = 1.0f;  // self-loop contributes 1
}

__global__ void k_accum_deg(const int* __restrict__ dst, float* __restrict__ deg, int E) {
    int e = blockIdx.x * blockDim.x + threadIdx.x;
    if (e < E) atomicAdd(&deg[dst[e]], 1.0f);
}

__global__ void k_dinv(float* __restrict__ deg, int N) {
    int i = blockIdx.x * blockDim.x + threadIdx.x;
    if (i < N) deg[i] = rsqrtf(deg[i]);  // deg >= 1 always (self loop)
}

// ---------------------------------------------------------------------------
// fp32 WMMA GEMM: H[N x FOUT] = X[N x FIN] * W[FIN x FOUT]
// One wave computes a full 16-row x FOUT strip: TC = FOUT/16 accumulator
// fragments, A fragment loaded once per K-step and reused across all TC
// V_WMMA_F32_16X16X4_F32 issues.
//
// fp32 A (16x4) layout : lane L holds row M = L%16; VGPR0 -> K = 2*(L>=16),
//                        VGPR1 -> K = 2*(L>=16)+1.
// fp32 B (4x16) layout : lane L holds col N = L%16; same K pattern as A.
// fp32 D (16x16) layout: VGPR v holds M = v + 8*(L>=16), N = L%16.
//
// For FOUT=8 (<16): B column address is clamped in-bounds; garbage columns of
// B only affect garbage (unstored) columns of D, so no masking is needed and
// EXEC stays all-ones around every WMMA.
// ---------------------------------------------------------------------------
template <int FIN, int FOUT>
__global__ void k_gemm_wmma_f32(const float* __restrict__ X,
                                const float* __restrict__ W,
                                float* __restrict__ H, int N) {
    constexpr int TC = (FOUT + 15) / 16;  // column tiles per strip
    const int tr = blockIdx.x * (blockDim.x >> 5) + (threadIdx.x >> 5);
    if (tr * 16 >= N) return;            // uniform per wave: EXEC stays full
    const int lane = threadIdx.x & 31;
    const int lo   = lane & 15;
    const int hi   = lane >> 4;          // 0 or 1

    const int arow = tr * 16 + lo;                       // A row this lane holds
    const int bcol = (lo < FOUT) ? lo : (FOUT - 1);      // clamped (FOUT=8 case)
    const float* __restrict__ xrow = X + (size_t)arow * (size_t)FIN;

    v8f acc[TC];
#pragma unroll
    for (int t = 0; t < TC; ++t) acc[t] = (v8f){};

    for (int k0 = 0; k0 < FIN; k0 += 4) {
        const int k = k0 + 2 * hi;
        v2f a = *(const v2f*)(xrow + k);                 // b64 load, 8B aligned
        const float* __restrict__ wr0 = W + (size_t)k * FOUT + bcol;
#pragma unroll
        for (int t = 0; t < TC; ++t) {
            v2f b;
            b.x = wr0[t * 16];
            b.y = wr0[t * 16 + FOUT];
            // 8 args: (neg_a, A, neg_b, B, c_mod, C, reuse_a, reuse_b)
            acc[t] = __builtin_amdgcn_wmma_f32_16x16x4_f32(
                false, a, false, b, (short)0, acc[t], false, false);
        }
    }

    if (lo < FOUT) {
#pragma unroll
        for (int t = 0; t < TC; ++t) {
#pragma unroll
            for (int v = 0; v < 8; ++v) {
                const int r = tr * 16 + v + 8 * hi;
                H[(size_t)r * FOUT + t * 16 + lo] = acc[t][v];
            }
        }
    }
}

// ---------------------------------------------------------------------------
// Aggregation: agg = D^-1/2 (A+I) D^-1/2 * H       (F is a power of two)
// ---------------------------------------------------------------------------
// Self-loop term initializes agg (no atomics, overwrites poison):
__global__ void k_selfloop_init(const float* __restrict__ H,
                                const float* __restrict__ dinv,
                                float* __restrict__ agg, long total, int logF) {
    long idx = (long)blockIdx.x * blockDim.x + threadIdx.x;
    if (idx >= total) return;
    int node = (int)(idx >> logF);
    float d = dinv[node];
    agg[idx] = H[idx] * d * d;
}

// Edge scatter: F consecutive lanes handle one edge -> coalesced gather of
// h[src] row and coalesced float atomics into agg[dst] row.
__global__ void k_edge_scatter(const float* __restrict__ H,
                               const float* __restrict__ dinv,
                               const int* __restrict__ src,
                               const int* __restrict__ dst,
                               float* __restrict__ agg, int E, int logF) {
    const int F  = 1 << logF;
    const int le = threadIdx.x >> logF;
    const int f  = threadIdx.x & (F - 1);
    const int e  = blockIdx.x * (blockDim.x >> logF) + le;
    if (e >= E) return;
    const int s = src[e];
    const int d = dst[e];
    const float w = dinv[s] * dinv[d];
    atomicAdd(&agg[((size_t)d << logF) + f], H[((size_t)s << logF) + f] * w);
}

// out = relu(agg + b)
__global__ void k_bias_relu(const float* __restrict__ agg,
                            const float* __restrict__ bias,
                            float* __restrict__ out, long total, int logF) {
    long idx = (long)blockIdx.x * blockDim.x + threadIdx.x;
    if (idx >= total) return;
    int f = (int)(idx & ((1 << logF) - 1));
    float v = agg[idx] + bias[f];
    out[idx] = v > 0.0f ? v : 0.0f;
}

// ---------------------------------------------------------------------------
// Launch
// ---------------------------------------------------------------------------
extern "C" void kernel_launch(void* const* d_in, const int* in_sizes, int n_in,
                              void* d_out, int out_size, void* d_ws, size_t ws_size,
                              hipStream_t stream) {
    const int IN_C = 128;
    const float* x  = (const float*)d_in[0];
    const int*   ei = (const int*)d_in[1];
    const int N = in_sizes[0] / IN_C;     // 100000
    const int E = in_sizes[1] / 2;        // 1600000
    const int* src = ei;                  // edge_index[0]
    const int* dst = ei + E;              // edge_index[1]

    const float* Ws[5] = {(const float*)d_in[2], (const float*)d_in[4],
                          (const float*)d_in[6], (const float*)d_in[8],
                          (const float*)d_in[10]};
    const float* Bs[5] = {(const float*)d_in[3], (const float*)d_in[5],
                          (const float*)d_in[7], (const float*)d_in[9],
                          (const float*)d_in[11]};
    const int FoutArr[5] = {128, 64, 32, 16, 8};
    const int LogFArr[5] = {7, 6, 5, 4, 3};

    // Workspace: [dinv: N] [hbuf: N*128] [abuf: N*128]
    float* dinv = (float*)d_ws;
    float* hbuf = dinv + N;                       // N*4 bytes keeps 16B align
    float* abuf = hbuf + (size_t)N * 128;

    const int T = 256;

    // --- normalization: deg -> dinv ---
    k_init_deg <<<(N + T - 1) / T, T, 0, stream>>>(dinv, N);
    k_accum_deg<<<(E + T - 1) / T, T, 0, stream>>>(dst, dinv, E);
    k_dinv     <<<(N + T - 1) / T, T, 0, stream>>>(dinv, N);

    // GEMM grid: one wave per 16-row strip, 8 waves (256 threads) per block
    const int strips = (N + 15) / 16;
    const int gblk   = (strips + 7) / 8;

    // --- 5 GCN layers ---
    const float* xin = x;
    for (int l = 0; l < 5; ++l) {
        const int Fout = FoutArr[l];
        const int logF = LogFArr[l];

        // GEMM: hbuf = xin @ W   (templated on shapes for full unrolling)
        switch (l) {
        case 0: k_gemm_wmma_f32<128, 128><<<gblk, T, 0, stream>>>(xin, Ws[0], hbuf, N); break;
        case 1: k_gemm_wmma_f32<128,  64><<<gblk, T, 0, stream>>>(xin, Ws[1], hbuf, N); break;
        case 2: k_gemm_wmma_f32< 64,  32><<<gblk, T, 0, stream>>>(xin, Ws[2], hbuf, N); break;
        case 3: k_gemm_wmma_f32< 32,  16><<<gblk, T, 0, stream>>>(xin, Ws[3], hbuf, N); break;
        case 4: k_gemm_wmma_f32< 16,   8><<<gblk, T, 0, stream>>>(xin, Ws[4], hbuf, N); break;
        }

        // agg = self-loop term (initializes buffer)
        const long tot = (long)N * Fout;
        k_selfloop_init<<<(int)((tot + T - 1) / T), T, 0, stream>>>(hbuf, dinv, abuf, tot, logF);

        // agg += scatter of normalized edge messages
        const int epb  = T >> logF;                 // edges per block
        const int sblk = (E + epb - 1) / epb;
        k_edge_scatter<<<sblk, T, 0, stream>>>(hbuf, dinv, src, dst, abuf, E, logF);

        // out = relu(agg + b); last layer writes d_out, else in place (abuf
        // becomes next layer's input)
        float* outp = (l == 4) ? (float*)d_out : abuf;
        k_bias_relu<<<(int)((tot + T - 1) / T), T, 0, stream>>>(abuf, Bs[l], outp, tot, logF);

        xin = abuf;
    }
}